// RelationalGraphletConvolution_26637387170236
// MI455X (gfx1250) — compile-verified
//
#include <hip/hip_runtime.h>

typedef __attribute__((ext_vector_type(2))) float v2f;
typedef __attribute__((ext_vector_type(8))) float v8f;

namespace {
constexpr int kNObj        = 32;                      // N objects
constexpr int kD           = 16;                      // relation dim
constexpr int kF           = 16;                      // filters
constexpr int kKTot        = 144;                     // 3*3*16 reduction length
constexpr int kGroups      = 4960;                    // C(32,3)
constexpr int kTilesTotal  = (32 * kGroups) / 16;     // 9920 M-tiles of 16 rows
constexpr int kWavesPerBlk = 8;                       // 256 threads, wave32
}

__global__ __launch_bounds__(256)
void rgc_wmma_f32_kernel(const float* __restrict__ R,
                         const float* __restrict__ filt,
                         float* __restrict__ out) {
  const int lane = threadIdx.x & 31;
  const int wave = threadIdx.x >> 5;
  const int tile = blockIdx.x * kWavesPerBlk + wave;   // 0..9919
  const int m    = lane & 15;                          // row within tile / filter col
  const int hi   = lane >> 4;                          // K-phase select (0 or 1)

  // ---- per-lane row identity: bg -> (b, g) -> lexicographic 3-combination ----
  const int bg = tile * 16 + m;
  const int b  = bg / kGroups;
  int r        = bg - b * kGroups;

  int c0 = 0;
  for (;;) {                                  // first element of combination
    const int cnt = ((31 - c0) * (30 - c0)) >> 1;   // C(31-c0, 2)
    if (r < cnt) break;
    r -= cnt; ++c0;
  }
  int c1 = c0 + 1;
  for (;;) {                                  // second element
    const int cnt = 31 - c1;
    if (r < cnt) break;
    r -= cnt; ++c1;
  }
  const int c2 = c1 + 1 + r;                  // third element

  // 9 base element-offsets into R: R[b, g_i, g_j, 0]
  int offs[9];
  {
    const int gi[3] = {c0, c1, c2};
#pragma unroll
    for (int i = 0; i < 3; ++i)
#pragma unroll
      for (int j = 0; j < 3; ++j)
        offs[i * 3 + j] = ((b * kNObj + gi[i]) * kNObj + gi[j]) * kD;
  }

  const float* frow = filt + m * kKTot;       // filters[f = lane%16, :] (row-major 144)
  const int dh = hi << 1;                     // +2 K-phase for lanes 16..31

  // ---- K loop: 36 steps of V_WMMA_F32_16X16X4_F32, two accumulators ----
  v8f acc0 = {};
  v8f acc1 = {};
#pragma unroll
  for (int s = 0; s < 36; ++s) {
    const int k0  = 4 * s;                    // compile-time per unrolled step
    const int seg = k0 >> 4;                  // which (i,j) 16-float block (0..8)
    const int d0  = (k0 & 15) + dh;           // d within block (+K phase)

    // A operand: R[b, g_i, g_j, d0..d0+1]  (8B, 8B-aligned)
    v2f a = *reinterpret_cast<const v2f*>(R + offs[seg] + d0);
    // B operand: filters[n, k0+dh .. k0+dh+1] (8B, 8B-aligned)
    v2f bm = *reinterpret_cast<const v2f*>(frow + k0 + dh);

    if (s & 1)
      acc1 = __builtin_amdgcn_wmma_f32_16x16x4_f32(false, a, false, bm,
                                                   (short)0, acc1, false, false);
    else
      acc0 = __builtin_amdgcn_wmma_f32_16x16x4_f32(false, a, false, bm,
                                                   (short)0, acc0, false, false);
  }

  // ---- store D tile: lane l, vgpr r -> out[(tile*16 + r + 8*hi)*16 + lane%16] ----
  float* orow = out + (tile * 16 + hi * 8) * kF + m;
#pragma unroll
  for (int rr = 0; rr < 8; ++rr) {
    orow[rr * kF] = acc0[rr] + acc1[rr];
  }
}

extern "C" void kernel_launch(void* const* d_in, const int* in_sizes, int n_in,
                              void* d_out, int out_size, void* d_ws, size_t ws_size,
                              hipStream_t stream) {
  (void)in_sizes; (void)n_in; (void)out_size; (void)d_ws; (void)ws_size;
  const float* R    = (const float*)d_in[0];   // (32,32,32,16) f32
  const float* filt = (const float*)d_in[1];   // (16,3,3,16)  f32
  float* out        = (float*)d_out;           // (32,4960,16) f32

  dim3 grid(kTilesTotal / kWavesPerBlk);       // 1240 blocks
  dim3 block(kWavesPerBlk * 32);               // 256 threads = 8 waves (wave32)
  rgc_wmma_f32_kernel<<<grid, block, 0, stream>>>(R, filt, out);
}